// VRAggregator_63385127354556
// MI455X (gfx1250) — compile-verified
//
#include <hip/hip_runtime.h>
#include <cstdint>

// ---------------------------------------------------------------------------
// VRAggregator for MI455X (gfx1250).
// Memory-bound scatter/gather workload: ~0.7 GFLOP vs ~3-4 GB traffic
// (AI < 0.1 FLOP/B) -> no WMMA; optimize the data-movement path instead:
//  * one wave32 per edge, each lane owns 8 floats (2 x float4) of the d=256 row
//  * gathered rows staged via GLOBAL_LOAD_ASYNC_TO_LDS_B128 + S_WAIT_ASYNCCNT
//    (gfx1250 async-memory path), falling back to direct b128 loads
//  * segment-sum via hardware global_atomic_add_f32 (L2-resident: the 64 MB
//    output fits easily in the 192 MB L2, so scatter RMW stays on-chip)
// ---------------------------------------------------------------------------

#define D   256
#define D4  64    // float4 chunks per row

#if defined(__HIP_DEVICE_COMPILE__) && \
    __has_builtin(__builtin_amdgcn_global_load_async_to_lds_b128) && \
    __has_builtin(__builtin_amdgcn_s_wait_asynccnt)
#define USE_ASYNC 1
#else
#define USE_ASYNC 0
#endif

// The builtin's parameters are int-vector pointers:
//   (v4i addrspace(1)*, v4i addrspace(3)*, imm offset, imm cpol)
typedef int vi4 __attribute__((vector_size(16)));
#define GAS(p) ((__attribute__((address_space(1))) vi4*)(p))
#define LAS(p) ((__attribute__((address_space(3))) vi4*)(p))

__device__ __forceinline__ void atomic_fadd(float* p, float v) {
  // lowers to global_atomic_add_f32 (no CAS loop)
  unsafeAtomicAdd(p, v);
}

#if USE_ASYNC
// Stage one 1KB row (256 floats) into LDS: each of 32 lanes issues two
// 16B async copies.  Caller must s_wait_asynccnt before reading.
__device__ __forceinline__ void stage_row_async(const float* g, float* l, int lane) {
  __builtin_amdgcn_global_load_async_to_lds_b128(GAS(g + (lane << 2)),
                                                 LAS(l + (lane << 2)), 0, 0);
  __builtin_amdgcn_global_load_async_to_lds_b128(GAS(g + ((lane + 32) << 2)),
                                                 LAS(l + ((lane + 32) << 2)), 0, 0);
}
#endif

// ---------------------------------------------------------------------------
// Kernel 1: mu_self / var_self copy + zero the neighbour accumulators.
// d_out layout (floats): [ mu_out (n_out x 512) | var_out (n_out x 512) ]
// ---------------------------------------------------------------------------
__global__ void __launch_bounds__(256)
vr_init(const float* __restrict__ mu, const float* __restrict__ var,
        float* __restrict__ out, int n_out) {
  int gid = blockIdx.x * blockDim.x + threadIdx.x;   // over n_out * D4
  if (gid >= n_out * D4) return;
  int r = gid >> 6;
  int j = gid & 63;
  const float4* mu4  = (const float4*)mu;
  const float4* var4 = (const float4*)var;
  float4* o = (float4*)out;
  size_t muBase  = (size_t)r * 128;                  // out row stride: 128 float4
  size_t varBase = (size_t)n_out * 128 + muBase;
  float4 z = make_float4(0.f, 0.f, 0.f, 0.f);
  o[muBase + j]       = mu4[(size_t)r * D4 + j];
  o[muBase + 64 + j]  = z;
  o[varBase + j]      = var4[(size_t)r * D4 + j];
  o[varBase + 64 + j] = z;
}

// ---------------------------------------------------------------------------
// Kernel 2: adj edges -> mu term  v*(mu[c]-muh[i])  and var term v^2*dsigma^2
// ---------------------------------------------------------------------------
__global__ void __launch_bounds__(256)
vr_adj(const float* __restrict__ mu, const float* __restrict__ var,
       const float* __restrict__ muh, const float* __restrict__ varh,
       const int* __restrict__ ifield,
       const float* __restrict__ vals, const int* __restrict__ rows,
       const int* __restrict__ cols,
       float* __restrict__ out, int nnz, int n_out) {
  int wave = (blockIdx.x * blockDim.x + threadIdx.x) >> 5;
  int lane = threadIdx.x & 31;
  if (wave >= nnz) return;

  int   c = cols[wave];
  int   r = rows[wave];
  float v = vals[wave];
  int   i = ifield[c];
  float v2 = v * v;

  const float* mrow  = mu   + (size_t)c * D;
  const float* srow  = var  + (size_t)c * D;
  const float* mhrow = muh  + (size_t)i * D;
  const float* shrow = varh + (size_t)i * D;
  float* omu = out + (size_t)r * (2 * D) + D;
  float* ova = out + (size_t)n_out * (2 * D) + (size_t)r * (2 * D) + D;

  const float *p_m, *p_s, *p_mh, *p_sh;
#if USE_ASYNC
  __shared__ __align__(16) float stage[8][4][D];     // 32 KB / block
  int w = threadIdx.x >> 5;
  stage_row_async(mrow,  &stage[w][0][0], lane);
  stage_row_async(srow,  &stage[w][1][0], lane);
  stage_row_async(mhrow, &stage[w][2][0], lane);
  stage_row_async(shrow, &stage[w][3][0], lane);
  __builtin_amdgcn_s_wait_asynccnt(0);
  p_m = &stage[w][0][0]; p_s = &stage[w][1][0];
  p_mh = &stage[w][2][0]; p_sh = &stage[w][3][0];
#else
  p_m = mrow; p_s = srow; p_mh = mhrow; p_sh = shrow;
#endif

#pragma unroll
  for (int h = 0; h < 2; ++h) {
    int k = (lane + h * 32) << 2;                    // float index in row
    float4 m  = *(const float4*)(p_m  + k);
    float4 s  = *(const float4*)(p_s  + k);
    float4 mh = *(const float4*)(p_mh + k);
    float4 sh = *(const float4*)(p_sh + k);
    float dm[4] = {m.x - mh.x, m.y - mh.y, m.z - mh.z, m.w - mh.w};
    float ds[4] = {__builtin_sqrtf(s.x) - __builtin_sqrtf(sh.x),
                   __builtin_sqrtf(s.y) - __builtin_sqrtf(sh.y),
                   __builtin_sqrtf(s.z) - __builtin_sqrtf(sh.z),
                   __builtin_sqrtf(s.w) - __builtin_sqrtf(sh.w)};
#pragma unroll
    for (int t = 0; t < 4; ++t) {
      atomic_fadd(omu + k + t, v * dm[t]);
      atomic_fadd(ova + k + t, v2 * ds[t] * ds[t]);
    }
  }
}

// ---------------------------------------------------------------------------
// Kernel 3: fadj edges -> mu term v*muh[ff[c]]  and var term v^2*varh[ff[c]]
// ---------------------------------------------------------------------------
__global__ void __launch_bounds__(256)
vr_fadj(const float* __restrict__ muh, const float* __restrict__ varh,
        const int* __restrict__ ffield,
        const float* __restrict__ vals, const int* __restrict__ rows,
        const int* __restrict__ cols,
        float* __restrict__ out, int nnz, int n_out) {
  int wave = (blockIdx.x * blockDim.x + threadIdx.x) >> 5;
  int lane = threadIdx.x & 31;
  if (wave >= nnz) return;

  int   c = cols[wave];
  int   r = rows[wave];
  float v = vals[wave];
  int   f = ffield[c];
  float v2 = v * v;

  const float* mbrow = muh  + (size_t)f * D;
  const float* vbrow = varh + (size_t)f * D;
  float* omu = out + (size_t)r * (2 * D) + D;
  float* ova = out + (size_t)n_out * (2 * D) + (size_t)r * (2 * D) + D;

  const float *p_mb, *p_vb;
#if USE_ASYNC
  __shared__ __align__(16) float stage[8][2][D];     // 16 KB / block
  int w = threadIdx.x >> 5;
  stage_row_async(mbrow, &stage[w][0][0], lane);
  stage_row_async(vbrow, &stage[w][1][0], lane);
  __builtin_amdgcn_s_wait_asynccnt(0);
  p_mb = &stage[w][0][0]; p_vb = &stage[w][1][0];
#else
  p_mb = mbrow; p_vb = vbrow;
#endif

#pragma unroll
  for (int h = 0; h < 2; ++h) {
    int k = (lane + h * 32) << 2;
    float4 mb = *(const float4*)(p_mb + k);
    float4 vb = *(const float4*)(p_vb + k);
    float mbv[4] = {mb.x, mb.y, mb.z, mb.w};
    float vbv[4] = {vb.x, vb.y, vb.z, vb.w};
#pragma unroll
    for (int t = 0; t < 4; ++t) {
      atomic_fadd(omu + k + t, v * mbv[t]);
      atomic_fadd(ova + k + t, v2 * vbv[t]);
    }
  }
}

// ---------------------------------------------------------------------------
// Kernel 4: madj edges -> var term 2*v*dsigma*sigma_bar
// ---------------------------------------------------------------------------
__global__ void __launch_bounds__(256)
vr_madj(const float* __restrict__ var, const float* __restrict__ varh,
        const int* __restrict__ ifield,
        const float* __restrict__ vals, const int* __restrict__ rows,
        const int* __restrict__ cols,
        float* __restrict__ out, int nnz, int n_out) {
  int wave = (blockIdx.x * blockDim.x + threadIdx.x) >> 5;
  int lane = threadIdx.x & 31;
  if (wave >= nnz) return;

  int   c = cols[wave];
  int   r = rows[wave];
  float v2 = 2.0f * vals[wave];
  int   i = ifield[c];

  const float* srow  = var  + (size_t)c * D;
  const float* shrow = varh + (size_t)i * D;
  float* ova = out + (size_t)n_out * (2 * D) + (size_t)r * (2 * D) + D;

  const float *p_s, *p_sh;
#if USE_ASYNC
  __shared__ __align__(16) float stage[8][2][D];     // 16 KB / block
  int w = threadIdx.x >> 5;
  stage_row_async(srow,  &stage[w][0][0], lane);
  stage_row_async(shrow, &stage[w][1][0], lane);
  __builtin_amdgcn_s_wait_asynccnt(0);
  p_s = &stage[w][0][0]; p_sh = &stage[w][1][0];
#else
  p_s = srow; p_sh = shrow;
#endif

#pragma unroll
  for (int h = 0; h < 2; ++h) {
    int k = (lane + h * 32) << 2;
    float4 s  = *(const float4*)(p_s  + k);
    float4 sh = *(const float4*)(p_sh + k);
    float sb[4] = {__builtin_sqrtf(sh.x), __builtin_sqrtf(sh.y),
                   __builtin_sqrtf(sh.z), __builtin_sqrtf(sh.w)};
    float ds[4] = {__builtin_sqrtf(s.x) - sb[0], __builtin_sqrtf(s.y) - sb[1],
                   __builtin_sqrtf(s.z) - sb[2], __builtin_sqrtf(s.w) - sb[3]};
#pragma unroll
    for (int t = 0; t < 4; ++t) {
      atomic_fadd(ova + k + t, v2 * ds[t] * sb[t]);
    }
  }
}

// ---------------------------------------------------------------------------
// Kernel 5: var_neighbour = relu(acc) + 1e-10
// ---------------------------------------------------------------------------
__global__ void __launch_bounds__(256)
vr_final(float* __restrict__ out, int n_out) {
  int gid = blockIdx.x * blockDim.x + threadIdx.x;   // over n_out * D4
  if (gid >= n_out * D4) return;
  int r = gid >> 6;
  int j = gid & 63;
  float4* p = (float4*)out + (size_t)n_out * 128 + (size_t)r * 128 + 64 + j;
  float4 x = *p;
  x.x = fmaxf(x.x, 0.f) + 1e-10f;
  x.y = fmaxf(x.y, 0.f) + 1e-10f;
  x.z = fmaxf(x.z, 0.f) + 1e-10f;
  x.w = fmaxf(x.w, 0.f) + 1e-10f;
  *p = x;
}

// ---------------------------------------------------------------------------
extern "C" void kernel_launch(void* const* d_in, const int* in_sizes, int n_in,
                              void* d_out, int out_size, void* d_ws, size_t ws_size,
                              hipStream_t stream) {
  const float* mu        = (const float*)d_in[0];
  const float* var       = (const float*)d_in[1];
  const float* muh       = (const float*)d_in[2];
  const float* varh      = (const float*)d_in[3];
  const float* adj_vals  = (const float*)d_in[4];
  const float* fadj_vals = (const float*)d_in[5];
  const float* madj_vals = (const float*)d_in[6];
  const int*   adj_rows  = (const int*)d_in[7];
  const int*   adj_cols  = (const int*)d_in[8];
  const int*   fadj_rows = (const int*)d_in[9];
  const int*   fadj_cols = (const int*)d_in[10];
  const int*   madj_rows = (const int*)d_in[11];
  const int*   madj_cols = (const int*)d_in[12];
  const int*   ifield    = (const int*)d_in[13];
  const int*   ffield    = (const int*)d_in[14];
  float* out = (float*)d_out;

  const int nnz   = in_sizes[4];
  const int n_out = out_size / (4 * D);   // out = 2 tensors x n_out x 2D floats

  const int ewThreads  = n_out * D4;
  const int ewBlocks   = (ewThreads + 255) / 256;
  const int edgeBlocks = (nnz * 32 + 255) / 256;     // one wave32 per edge

  vr_init<<<ewBlocks, 256, 0, stream>>>(mu, var, out, n_out);
  vr_adj<<<edgeBlocks, 256, 0, stream>>>(mu, var, muh, varh, ifield,
                                         adj_vals, adj_rows, adj_cols,
                                         out, nnz, n_out);
  vr_fadj<<<edgeBlocks, 256, 0, stream>>>(muh, varh, ffield,
                                          fadj_vals, fadj_rows, fadj_cols,
                                          out, nnz, n_out);
  vr_madj<<<edgeBlocks, 256, 0, stream>>>(var, varh, ifield,
                                          madj_vals, madj_rows, madj_cols,
                                          out, nnz, n_out);
  vr_final<<<ewBlocks, 256, 0, stream>>>(out, n_out);
  (void)d_ws; (void)ws_size; (void)n_in;
}